// AdaptiveDistillationLoss_41034117546380
// MI455X (gfx1250) — compile-verified
//
#include <hip/hip_runtime.h>
#include <hip/hip_bf16.h>

// ---- vector types -----------------------------------------------------------
typedef float f32x4 __attribute__((ext_vector_type(4)));
typedef int   i32x4 __attribute__((ext_vector_type(4)));
typedef float v2f   __attribute__((ext_vector_type(2)));
typedef float v8f   __attribute__((ext_vector_type(8)));

#define NBLK 2048
#define NTHR 256

// ---- exact 32-lane sum via V_WMMA_F32_16X16X4_F32 ---------------------------
// A: lane L contributes its partial at K-column 0 (a.x), zeros elsewhere.
// B: all-ones 4x16.  D[m][n] = sum_k A[m][k]*1 = p_m + p_{m+16} (lane-pair sum).
// Lane n (0..15) holds D[0..7][n] in c[0..7]; lane n (16..31) holds D[8..15][.].
// Sum of the 8 accumulator regs gives lanes 0-15: sum(p0..p7,p16..p23),
// lanes 16-31: sum(p8..p15,p24..p31); one xor-16 shuffle completes the total.
__device__ __forceinline__ float wave_sum_wmma(float v) {
  v2f a; a.x = v;    a.y = 0.0f;
  v2f b; b.x = 1.0f; b.y = 1.0f;
  v8f c = {};
  c = __builtin_amdgcn_wmma_f32_16x16x4_f32(
      /*neg_a=*/false, a, /*neg_b=*/false, b,
      /*c_mod=*/(short)0, c, /*reuse_a=*/false, /*reuse_b=*/false);
  float s = ((c[0] + c[1]) + (c[2] + c[3])) + ((c[4] + c[5]) + (c[6] + c[7]));
  s += __shfl_xor(s, 16, 32);
  return s;
}

// ---- per-sample fused loss: 0.5*KL + 0.5*CE ---------------------------------
__device__ __forceinline__ float sample_loss(float la, float lb, float lc,
                                             float ta, float tb, float tc,
                                             float cf, int y) {
  // adaptive temperature
  float low  = fminf(fmaf(0.6f - cf, 2.0f, 2.5f), 3.0f);
  float temp = cf > 0.9f ? 1.5f : (cf > 0.6f ? 2.0f : low);
  float it   = 1.0f / temp;

  // KL(teacher || softmax(logits/T)) = sum t*ln t - sum t*s + lse(s)*sum t
  float sa = la * it, sb = lb * it, sc = lc * it;
  float m1   = fmaxf(sa, fmaxf(sb, sc));
  float lse1 = m1 + __logf(__expf(sa - m1) + __expf(sb - m1) + __expf(sc - m1));
  float sumT = (ta + tb) + tc;
  float tlt  = fmaf(ta, __logf(ta), fmaf(tb, __logf(tb), tc * __logf(tc)));
  float tdots= fmaf(ta, sa, fmaf(tb, sb, tc * sc));
  float kl   = tlt - tdots + lse1 * sumT;

  // CE = lse(logits) - logits[y]
  float m2   = fmaxf(la, fmaxf(lb, lc));
  float lse2 = m2 + __logf(__expf(la - m2) + __expf(lb - m2) + __expf(lc - m2));
  float ly   = (y == 0) ? la : ((y == 1) ? lb : lc);

  return 0.5f * (kl + (lse2 - ly));   // ALPHA = 0.5
}

// ---- pass 1: streaming partial sums (one float per block) -------------------
__global__ void __launch_bounds__(NTHR)
partial_kernel(const float* __restrict__ logits,
               const int*   __restrict__ labels,
               const float* __restrict__ soft,
               const float* __restrict__ conf,
               float* __restrict__ partials, int n) {
  const size_t tid      = (size_t)blockIdx.x * blockDim.x + threadIdx.x;
  const size_t nthreads = (size_t)gridDim.x * blockDim.x;
  const size_t ngrp     = (size_t)n >> 2;

  float acc = 0.0f;
  for (size_t g = tid; g < ngrp; g += nthreads) {
    const f32x4* lp = (const f32x4*)(logits + 12 * g);   // 3x float4, 16B aligned
    const f32x4* sp = (const f32x4*)(soft   + 12 * g);
    f32x4 l0 = __builtin_nontemporal_load(lp);
    f32x4 l1 = __builtin_nontemporal_load(lp + 1);
    f32x4 l2 = __builtin_nontemporal_load(lp + 2);
    f32x4 s0 = __builtin_nontemporal_load(sp);
    f32x4 s1 = __builtin_nontemporal_load(sp + 1);
    f32x4 s2 = __builtin_nontemporal_load(sp + 2);
    f32x4 cf = __builtin_nontemporal_load((const f32x4*)(conf   + 4 * g));
    i32x4 yb = __builtin_nontemporal_load((const i32x4*)(labels + 4 * g));

    acc += sample_loss(l0.x, l0.y, l0.z, s0.x, s0.y, s0.z, cf.x, yb.x);
    acc += sample_loss(l0.w, l1.x, l1.y, s0.w, s1.x, s1.y, cf.y, yb.y);
    acc += sample_loss(l1.z, l1.w, l2.x, s1.z, s1.w, s2.x, cf.z, yb.z);
    acc += sample_loss(l2.y, l2.z, l2.w, s2.y, s2.z, s2.w, cf.w, yb.w);
  }
  // scalar tail (dead for n % 4 == 0, kept for generality; converges before WMMA)
  for (size_t i = (ngrp << 2) + tid; i < (size_t)n; i += nthreads) {
    acc += sample_loss(logits[3 * i], logits[3 * i + 1], logits[3 * i + 2],
                       soft[3 * i],   soft[3 * i + 1],   soft[3 * i + 2],
                       conf[i], labels[i]);
  }

  // block reduction: WMMA wave sum -> LDS across 8 waves -> one partial
  float ws = wave_sum_wmma(acc);
  __shared__ float lds[NTHR / 32];
  const int lane = threadIdx.x & 31, wave = threadIdx.x >> 5;
  if (lane == 0) lds[wave] = ws;
  __syncthreads();
  if (threadIdx.x == 0) {
    float s = 0.0f;
#pragma unroll
    for (int w = 0; w < NTHR / 32; ++w) s += lds[w];
    partials[blockIdx.x] = s;
  }
}

// ---- pass 2: deterministic final reduction in double ------------------------
__global__ void __launch_bounds__(NTHR)
final_kernel(const float* __restrict__ partials, int nPartials,
             float* __restrict__ out, int n) {
  __shared__ double lds[NTHR];
  double s = 0.0;
  for (int i = threadIdx.x; i < nPartials; i += NTHR) s += (double)partials[i];
  lds[threadIdx.x] = s;
  __syncthreads();
  if (threadIdx.x == 0) {
    double t = 0.0;
#pragma unroll 8
    for (int i = 0; i < NTHR; ++i) t += lds[i];
    out[0] = (float)(t / (double)n);
  }
}

extern "C" void kernel_launch(void* const* d_in, const int* in_sizes, int n_in,
                              void* d_out, int out_size, void* d_ws, size_t ws_size,
                              hipStream_t stream) {
  const float* logits = (const float*)d_in[0];   // (B, 3) f32
  const int*   labels = (const int*)d_in[1];     // (B,)   i32
  const float* soft   = (const float*)d_in[2];   // (B, 3) f32
  const float* conf   = (const float*)d_in[3];   // (B,)   f32
  float*       out    = (float*)d_out;
  float*       part   = (float*)d_ws;            // NBLK floats = 8 KB scratch
  const int    n      = in_sizes[1];             // B

  partial_kernel<<<NBLK, NTHR, 0, stream>>>(logits, labels, soft, conf, part, n);
  final_kernel<<<1, NTHR, 0, stream>>>(part, NBLK, out, n);
}